// BlockSparseLinear_20899310862698
// MI455X (gfx1250) — compile-verified
//
#include <hip/hip_runtime.h>

typedef __attribute__((ext_vector_type(2))) float v2f;
typedef __attribute__((ext_vector_type(4))) float v4f;
typedef __attribute__((ext_vector_type(8))) float v8f;

#define M_TOT 8192
#define N_TOT 4096
#define K_TOT 4096

#define MT 128          // workgroup M tile
#define NT 128          // workgroup N tile
#define KT 32           // K slab staged in LDS per iteration
#define LDSTR 34        // padded row stride (floats): 8B-aligned b64s, disjoint bank pairs
#define A_ELEMS (MT * LDSTR)
#define SLAB_ELEMS (2 * A_ELEMS)   // A tile + B tile per buffer

// Stage one 128x32 slab of X and W into LDS (per-thread: 4 float4 loads each).
__device__ __forceinline__ void stage_slab(const float* __restrict__ xp,
                                           const float* __restrict__ wp,
                                           float* __restrict__ As,
                                           float* __restrict__ Bs,
                                           int sr, int sc) {
    #pragma unroll
    for (int p = 0; p < 4; ++p) {
        const size_t goff = (size_t)(32 * p) * K_TOT;
        v4f a = *(const v4f*)(xp + goff);
        v4f b = *(const v4f*)(wp + goff);
        const int l = (sr + 32 * p) * LDSTR + sc;
        *(v2f*)&As[l]     = a.lo;
        *(v2f*)&As[l + 2] = a.hi;
        *(v2f*)&Bs[l]     = b.lo;
        *(v2f*)&Bs[l + 2] = b.hi;
    }
}

__global__ __launch_bounds__(256) void bs_linear_wmma_f32(
    const float* __restrict__ X,      // [8192][4096]
    const float* __restrict__ W,      // [4096][4096] row = out-feature
    const float* __restrict__ bias,   // [4096]
    float* __restrict__ Y)            // [8192][4096]
{
    __shared__ float lds[2][SLAB_ELEMS];   // double-buffered A+B slabs (~68 KB)

    const int tileN = blockIdx.x;          // 0..31
    const int tileM = blockIdx.y;          // 0..63
    const int mBase = tileM * MT;
    const int nBase = tileN * NT;

    const int tid  = threadIdx.x;
    const int wave = tid >> 5;             // 0..7
    const int lane = tid & 31;
    const int lid  = lane & 15;
    const int half = lane >> 4;            // 0: K={0,1}/M=r ; 1: K={2,3}/M=r+8
    const int wm   = wave >> 2;            // 0..1 -> 64 M rows each
    const int wn   = wave & 3;             // 0..3 -> 32 N cols each

    // staging coords: 128 rows x 32 cols; thread covers (sr + 32p, sc..sc+3)
    const int sr = tid >> 3;               // 0..31
    const int sc = (tid & 7) * 4;          // 0,4,...,28

    const float* xp = X + (size_t)(mBase + sr) * K_TOT + sc;
    const float* wp = W + (size_t)(nBase + sr) * K_TOT + sc;

    v8f acc[4][2];
    #pragma unroll
    for (int i = 0; i < 4; ++i)
        #pragma unroll
        for (int j = 0; j < 2; ++j)
            acc[i][j] = (v8f)0.0f;

    // prologue: fill buffer 0
    stage_slab(xp, wp, lds[0], lds[0] + A_ELEMS, sr, sc);
    __syncthreads();

    int buf = 0;
    for (int k0 = 0; k0 < K_TOT; k0 += KT) {
        // stage next slab into the other buffer while computing this one
        if (k0 + KT < K_TOT)
            stage_slab(xp + (k0 + KT), wp + (k0 + KT),
                       lds[buf ^ 1], lds[buf ^ 1] + A_ELEMS, sr, sc);

        const float* As = lds[buf];
        const float* Bs = lds[buf] + A_ELEMS;

        #pragma unroll
        for (int kk = 0; kk < KT; kk += 4) {
            v2f afrag[4];
            v2f bfrag[2];
            #pragma unroll
            for (int tm = 0; tm < 4; ++tm)
                afrag[tm] = *(const v2f*)&As[(wm * 64 + tm * 16 + lid) * LDSTR + kk + 2 * half];
            #pragma unroll
            for (int tn = 0; tn < 2; ++tn)
                bfrag[tn] = *(const v2f*)&Bs[(wn * 32 + tn * 16 + lid) * LDSTR + kk + 2 * half];

            #pragma unroll
            for (int tm = 0; tm < 4; ++tm)
                #pragma unroll
                for (int tn = 0; tn < 2; ++tn)
                    acc[tm][tn] = __builtin_amdgcn_wmma_f32_16x16x4_f32(
                        /*neg_a=*/false, afrag[tm],
                        /*neg_b=*/false, bfrag[tn],
                        /*c_mod=*/(short)0, acc[tm][tn],
                        /*reuse_a=*/false, /*reuse_b=*/false);
        }
        __syncthreads();
        buf ^= 1;
    }

    // epilogue: D layout -> VGPR r holds M = r + 8*half, N = lid
    #pragma unroll
    for (int tn = 0; tn < 2; ++tn) {
        const int col = nBase + wn * 32 + tn * 16 + lid;
        const float bv = bias[col];
        #pragma unroll
        for (int tm = 0; tm < 4; ++tm) {
            const int rbase = mBase + wm * 64 + tm * 16 + 8 * half;
            #pragma unroll
            for (int r = 0; r < 8; ++r) {
                Y[(size_t)(rbase + r) * N_TOT + col] = acc[tm][tn][r] + bv;
            }
        }
    }
}

extern "C" void kernel_launch(void* const* d_in, const int* in_sizes, int n_in,
                              void* d_out, int out_size, void* d_ws, size_t ws_size,
                              hipStream_t stream) {
    const float* x    = (const float*)d_in[0];   // 8192*4096
    const float* w    = (const float*)d_in[1];   // 4096*4096 (block-sparse masked)
    const float* bias = (const float*)d_in[2];   // 4096
    float* y          = (float*)d_out;           // 8192*4096

    dim3 grid(N_TOT / NT, M_TOT / MT);           // (32, 64)
    dim3 block(256);
    bs_linear_wmma_f32<<<grid, block, 0, stream>>>(x, w, bias, y);
}